// RecurrentOjaUpdateRule_84859963834865
// MI455X (gfx1250) — compile-verified
//
#include <hip/hip_runtime.h>
#include <hip/hip_bf16.h>

// ---------------------------------------------------------------------------
// RecurrentOjaUpdateRule on MI455X (gfx1250, wave32, WMMA)
//
// Pipeline (all fp32, full precision — GEMM cost sits below the mandatory
// 1 GB W-stream bandwidth floor, so f32 WMMA is the optimal precision):
//   k_gemm1_ln : h_ln = LayerNorm(x @ ip_w + ip_b) * ln_g + ln_b   (fused)
//   k_gemm2    : h2   = [h_ln | z] @ mg_w + mg_b   (two K=512 passes)
//   k_gemm3    : kvb  = h2 @ kvb_w + kvb_b         (N=1032, edge-guarded)
//   k_oja      : per (b,h): ks' = softmax(ks - vs^T W)*sigmoid(lr);
//                out = vs (outer) ks' + W          (W read once via registers)
// ---------------------------------------------------------------------------

typedef float v2f __attribute__((ext_vector_type(2)));
typedef float v8f __attribute__((ext_vector_type(8)));

#define B_ROWS   4096
#define LAT      512
#define NH       8
#define DD       64
#define KVB_N    1032   // H*(2D+1)
#define LN_EPSF  1e-5f

// ---------------------------------------------------------------------------
// Wave-level 16x64 fp32 GEMM strip using V_WMMA_F32_16X16X4_F32.
// A fragment (16x4, ISA 7.12.2): lane l<16 holds A[l][k], A[l][k+1];
// lane l+16 holds A[l][k+2], A[l][k+3]  (K LSB -> VGPR, K MSB -> lane half).
// B fragment (4x16): row striped across lanes within a VGPR.
// Accumulators: acc[j][i] = C[M = i + 8*(lane>>4)][N = j*16 + (lane&15)].
// ---------------------------------------------------------------------------
__device__ __forceinline__ void wmma_strip(const float* __restrict__ A, int lda,
                                           const float* __restrict__ Bw, int ldb,
                                           int K, int rowBase, int colBase,
                                           int Nreal, v8f acc[4]) {
  const int lane = threadIdx.x & 31;
  const int half = lane >> 4;     // 0/1
  const int l15  = lane & 15;
  const float* arow = A + (size_t)(rowBase + l15) * lda;
  for (int k = 0; k < K; k += 4) {
    v2f a;
    const int ka = k + 2 * half;
    a.x = arow[ka];
    a.y = arow[ka + 1];
#pragma unroll
    for (int j = 0; j < 4; ++j) {
      const int col  = colBase + j * 16 + l15;
      const int colc = (col < Nreal) ? col : (Nreal - 1);
      float b0 = Bw[(size_t)(k + half) * ldb + colc];
      float b1 = Bw[(size_t)(k + 2 + half) * ldb + colc];
      b0 = (col < Nreal) ? b0 : 0.0f;
      b1 = (col < Nreal) ? b1 : 0.0f;
      v2f bf; bf.x = b0; bf.y = b1;
      acc[j] = __builtin_amdgcn_wmma_f32_16x16x4_f32(
          /*neg_a=*/false, a, /*neg_b=*/false, bf,
          /*c_mod=*/(short)0, acc[j], /*reuse_a=*/false, /*reuse_b=*/false);
    }
  }
}

__device__ __forceinline__ void store_strip(float* __restrict__ C, int ldc,
                                            const float* __restrict__ bias,
                                            int rowBase, int colBase, int Nreal,
                                            const v8f acc[4]) {
  const int lane = threadIdx.x & 31;
  const int half = lane >> 4;
  const int l15  = lane & 15;
#pragma unroll
  for (int j = 0; j < 4; ++j) {
    const int col = colBase + j * 16 + l15;
    if (col < Nreal) {
      const float bv = bias[col];
#pragma unroll
      for (int i = 0; i < 8; ++i) {
        const int row = rowBase + i + 8 * half;
        C[(size_t)row * ldc + col] = acc[j][i] + bv;
      }
    }
  }
}

// ------------------------- GEMM1 + fused LayerNorm -------------------------
// grid = 256 blocks (16 rows each), block = 256 threads = 8 waves (64 cols ea)
__global__ void __launch_bounds__(256)
k_gemm1_ln(const float* __restrict__ x, const float* __restrict__ ip_w,
           const float* __restrict__ ip_b, const float* __restrict__ ln_g,
           const float* __restrict__ ln_b, float* __restrict__ h_ln) {
  __shared__ float tile[16][LAT];
  __shared__ float red[16][16];
  __shared__ float mu[16], rsg[16];

  const int tid     = threadIdx.x;
  const int wave    = tid >> 5;
  const int rowBase = blockIdx.x * 16;
  const int colBase = wave * 64;

  v8f acc[4] = {};
  wmma_strip(x, LAT, ip_w, LAT, LAT, rowBase, colBase, LAT, acc);

  // Accumulators (+bias) into LDS tile for the cross-column LayerNorm.
  {
    const int lane = tid & 31, half = lane >> 4, l15 = lane & 15;
#pragma unroll
    for (int j = 0; j < 4; ++j) {
      const int col = colBase + j * 16 + l15;
      const float bv = ip_b[col];
#pragma unroll
      for (int i = 0; i < 8; ++i) tile[i + 8 * half][col] = acc[j][i] + bv;
    }
  }
  __syncthreads();

  // LayerNorm: 16 threads per row, 32 columns each.
  const int r  = tid >> 4;
  const int c0 = (tid & 15) * 32;
  float s = 0.0f;
  for (int c = 0; c < 32; ++c) s += tile[r][c0 + c];
  red[r][tid & 15] = s;
  __syncthreads();
  if ((tid & 15) == 0) {
    float t = 0.0f;
    for (int i = 0; i < 16; ++i) t += red[r][i];
    mu[r] = t * (1.0f / (float)LAT);
  }
  __syncthreads();
  const float m = mu[r];
  s = 0.0f;
  for (int c = 0; c < 32; ++c) {
    const float d = tile[r][c0 + c] - m;
    s += d * d;
  }
  red[r][tid & 15] = s;
  __syncthreads();
  if ((tid & 15) == 0) {
    float t = 0.0f;
    for (int i = 0; i < 16; ++i) t += red[r][i];
    rsg[r] = rsqrtf(t * (1.0f / (float)LAT) + LN_EPSF);
  }
  __syncthreads();
  const float rr = rsg[r];
  for (int c = 0; c < 32; ++c) {
    const int cc = c0 + c;
    h_ln[(size_t)(rowBase + r) * LAT + cc] =
        (tile[r][cc] - m) * rr * ln_g[cc] + ln_b[cc];
  }
}

// ------------------------- GEMM2: [h_ln | z] @ mg_w ------------------------
__global__ void __launch_bounds__(256)
k_gemm2(const float* __restrict__ h_ln, const float* __restrict__ z,
        const float* __restrict__ mg_w, const float* __restrict__ mg_b,
        float* __restrict__ h2) {
  const int wave    = threadIdx.x >> 5;
  const int rowBase = blockIdx.x * 16;
  const int colBase = wave * 64;
  v8f acc[4] = {};
  wmma_strip(h_ln, LAT, mg_w,             LAT, LAT, rowBase, colBase, LAT, acc);
  wmma_strip(z,    LAT, mg_w + LAT * LAT, LAT, LAT, rowBase, colBase, LAT, acc);
  store_strip(h2, LAT, mg_b, rowBase, colBase, LAT, acc);
}

// ------------------------- GEMM3: h2 @ kvb_w (N=1032) ----------------------
// 256 row-tiles x 17 col-strips = 4352 wave jobs = 544 blocks * 8 waves.
__global__ void __launch_bounds__(256)
k_gemm3(const float* __restrict__ h2, const float* __restrict__ kvb_w,
        const float* __restrict__ kvb_b, float* __restrict__ kvb) {
  const int wave = threadIdx.x >> 5;
  const int wg   = blockIdx.x * 8 + wave;
  const int rowTile = wg / 17;
  const int strip   = wg % 17;
  v8f acc[4] = {};
  wmma_strip(h2, LAT, kvb_w, KVB_N, LAT, rowTile * 16, strip * 64, KVB_N, acc);
  store_strip(kvb, KVB_N, kvb_b, rowTile * 16, strip * 64, KVB_N, acc);
}

// ------------------------- Oja epilogue ------------------------------------
// One 64-thread group per (b,h); thread e owns W column e in registers, so
// the 512 MB W tensor is read from HBM exactly once (bandwidth floor).
__global__ void __launch_bounds__(256)
k_oja(const float* __restrict__ Wm, const float* __restrict__ kvb,
      float* __restrict__ out) {
  __shared__ float vsb[4][DD];
  __shared__ float redb[4][DD];

  const int tid = threadIdx.x;
  const int p   = tid >> 6;          // pair slot within block
  const int e   = tid & 63;          // column index
  const long pair = (long)blockIdx.x * 4 + p;      // 0 .. 32767
  const int b = (int)(pair >> 3);
  const int h = (int)(pair & 7);

  const float* row = kvb + (size_t)b * KVB_N + h * (2 * DD + 1);
  const float* Wp  = Wm + (size_t)pair * (DD * DD);

  __builtin_prefetch(Wp + (size_t)e * 16, 0, 0);   // global_prefetch_b8 hint

  const float ks_raw = row[e];
  const float vs_e   = tanhf(row[DD + e]);
  const float lr     = 1.0f / (1.0f + __expf(-row[2 * DD]));

  vsb[p][e] = vs_e;
  __syncthreads();

  // Stream W column e into registers; accumulate ks_remove[e] = sum_d W[d][e]*vs[d]
  float wcol[DD];
  float accr = 0.0f;
#pragma unroll
  for (int d = 0; d < DD; ++d) {
    wcol[d] = Wp[d * DD + e];        // coalesced across e
    accr += wcol[d] * vsb[p][d];
  }
  const float s = ks_raw - accr;

  // softmax over the 64-wide group (2 waves) via LDS tree reductions
  redb[p][e] = s;
  __syncthreads();
  for (int off = 32; off >= 1; off >>= 1) {
    if (e < off) redb[p][e] = fmaxf(redb[p][e], redb[p][e + off]);
    __syncthreads();
  }
  const float mx = redb[p][0];
  __syncthreads();
  const float ex = __expf(s - mx);
  redb[p][e] = ex;
  __syncthreads();
  for (int off = 32; off >= 1; off >>= 1) {
    if (e < off) redb[p][e] += redb[p][e + off];
    __syncthreads();
  }
  const float ksf = ex / redb[p][0] * lr;

  // out[d][e] = vs[d] * ks'[e] + W[d][e]   (W from registers, write coalesced)
  float* op = out + (size_t)pair * (DD * DD);
#pragma unroll
  for (int d = 0; d < DD; ++d) op[d * DD + e] = vsb[p][d] * ksf + wcol[d];
}

// ---------------------------------------------------------------------------
extern "C" void kernel_launch(void* const* d_in, const int* in_sizes, int n_in,
                              void* d_out, int out_size, void* d_ws, size_t ws_size,
                              hipStream_t stream) {
  const float* x     = (const float*)d_in[0];
  const float* z     = (const float*)d_in[1];
  const float* W     = (const float*)d_in[2];
  const float* ip_w  = (const float*)d_in[3];
  const float* ip_b  = (const float*)d_in[4];
  const float* ln_g  = (const float*)d_in[5];
  const float* ln_b  = (const float*)d_in[6];
  const float* mg_w  = (const float*)d_in[7];
  const float* mg_b  = (const float*)d_in[8];
  const float* kvb_w = (const float*)d_in[9];
  const float* kvb_b = (const float*)d_in[10];
  float* out = (float*)d_out;

  float* ws   = (float*)d_ws;
  float* h_ln = ws;                                   // 4096*512
  float* h2   = h_ln + (size_t)B_ROWS * LAT;          // 4096*512
  float* kvb  = h2 + (size_t)B_ROWS * LAT;            // 4096*1032

  k_gemm1_ln<<<B_ROWS / 16, 256, 0, stream>>>(x, ip_w, ip_b, ln_g, ln_b, h_ln);
  k_gemm2<<<B_ROWS / 16, 256, 0, stream>>>(h_ln, z, mg_w, mg_b, h2);
  k_gemm3<<<(B_ROWS / 16) * 17 / 8, 256, 0, stream>>>(h2, kvb_w, kvb_b, kvb);
  k_oja<<<(B_ROWS * NH) / 4, 256, 0, stream>>>(W, kvb, out);
}